// Conv2D_45354854646009
// MI455X (gfx1250) — compile-verified
//
#include <hip/hip_runtime.h>
#include <stdint.h>

// CDNA5 / gfx1250: fp32 3x3 VALID conv.
//  - Roofline: 1.21 GFLOP vs 537 MB moved => 2.25 FLOP/byte, memory bound
//    (~23us floor at 23.3 TB/s). Design goal: one HBM pass, coalesced I/O.
//  - Tile staging: Tensor Data Mover (tensor_load_to_lds) DMAs a 34x66 fp32
//    tile (incl. halo) into LDS per block; TDM zero-fills out-of-tensor reads,
//    which implements the edge halo for free. Tracked via TENSORcnt.
//  - Compute: V_WMMA_F32_16X16X4_F32. A = 3x3 weights replicated down M
//    (taps 4i..4i+3 per WMMA, zero-padded to 12), B = im2col patches with
//    N = 16 output pixels; three chained WMMAs accumulate the 9-tap sum,
//    every row of D equals out[n], lane n stores acc[0].

typedef __attribute__((ext_vector_type(2))) float v2f;
typedef __attribute__((ext_vector_type(8))) float v8f;
typedef __attribute__((ext_vector_type(4))) unsigned int v4u;
typedef __attribute__((ext_vector_type(8))) unsigned int v8u;

#define W_IN  8192
#define H_IN  8192
#define W_OUT 8190
#define H_OUT 8190
#define TBX   64            // output cols per block
#define TBY   32            // output rows per block
#define LDSH  (TBY + 2)     // 34 rows incl. halo
#define LDSW  (TBX + 2)     // 66 cols incl. halo (TDM rows packed contiguously)

__global__ __launch_bounds__(256) void conv3x3_wmma_f32(
    const float* __restrict__ X,
    const float* __restrict__ Wt,
    const float* __restrict__ Bi,
    float* __restrict__ out)
{
    __shared__ float tile[LDSH * LDSW];
    __shared__ float wlds[12];

    const int tid  = threadIdx.x;
    const int lane = tid & 31;
    const int wave = tid >> 5;
    const int gx0  = blockIdx.x * TBX;
    const int gy0  = blockIdx.y * TBY;

    // stage 3x3 weights (zero-padded to 12 taps: K=4 x 3 WMMAs)
    if (tid < 12) wlds[tid] = (tid < 9) ? Wt[tid] : 0.0f;

    // ---- TDM: DMA the input tile (with halo) into LDS, wave 0 issues ----
    if (wave == 0) {
        // D# group 0: count=1 | lds_addr | global_addr(57b) | type=2
        uint64_t gaddr = (uint64_t)(uintptr_t)(X + (size_t)gy0 * W_IN + gx0);
        uint32_t laddr = (uint32_t)(uintptr_t)(void*)tile;  // LDS byte offset
        v4u g0;
        g0.x = 0x1u;                                   // count=1, user mode
        g0.y = laddr;                                  // lds_addr [63:32]
        g0.z = (uint32_t)(gaddr & 0xFFFFFFFFu);        // global_addr[31:0]
        g0.w = (uint32_t)((gaddr >> 32) & 0x1FFFFFFu)  // global_addr[56:32]
             | 0x80000000u;                            // type=2 ("image")

        // D# group 1: data_size=4B; tensor dims = remaining extent from tile
        // origin (TDM zero-fills reads past them -> halo zero padding);
        // tile = 66 x 34; row stride = 8192 elements.
        uint32_t td0 = (uint32_t)(W_IN - gx0);         // remaining cols
        uint32_t td1 = (uint32_t)(H_IN - gy0);         // remaining rows
        v8u g1;
        g1[0] = (2u << 16);                            // data_size=2 (4 bytes)
        g1[1] = (td0 & 0xFFFFu) << 16;                 // tensor_dim0[15:0]
        g1[2] = (td0 >> 16) | ((td1 & 0xFFFFu) << 16); // td0[31:16] | td1[15:0]
        g1[3] = (td1 >> 16) | ((uint32_t)LDSW << 16);  // td1[31:16] | tile_dim0
        g1[4] = (uint32_t)LDSH;                        // tile_dim1 | tile_dim2=0
        g1[5] = (uint32_t)W_IN;                        // tensor_dim0_stride[31:0]
        g1[6] = 0u;                                    // stride[47:32] | ts1[15:0]
        g1[7] = 0u;                                    // tensor_dim1_stride hi

        asm volatile("tensor_load_to_lds %0, %1"
                     :
                     : "s"(g0), "s"(g1)
                     : "memory");
        __builtin_amdgcn_s_wait_tensorcnt(0);
    }
    __syncthreads();

    // wave -> 16-wide column strip, 16-row half of the block tile
    const int wx  = wave & 3;
    const int wy  = wave >> 2;
    const int lxp = wx * 16;
    const int ly  = wy * 16;

    const int n    = lane & 15;       // output pixel within strip (B's N index)
    const int half = lane >> 4;       // lane half selects tap pair (K pair)

    // Per-lane fixed taps: WMMA i covers taps 4i..4i+3.
    // A 16x4 layout: lanes 0-15 hold K={0,1}, lanes 16-31 hold K={2,3};
    // B 4x16 rows are striped across lanes the same way (row K over N lanes).
    int off[3][2];
    v2f aw[3];
#pragma unroll
    for (int i = 0; i < 3; ++i) {
        int t0 = i * 4 + half * 2;
        int t1 = t0 + 1;
        int c0 = (t0 < 9) ? t0 : 0;   // clamp pad taps to a valid LDS offset
        int c1 = (t1 < 9) ? t1 : 0;   // (their weight is 0 so value is moot)
        int dy0 = c0 / 3, dx0 = c0 - dy0 * 3;
        int dy1 = c1 / 3, dx1 = c1 - dy1 * 3;
        off[i][0] = dy0 * LDSW + dx0 + n;
        off[i][1] = dy1 * LDSW + dx1 + n;
        aw[i].x = wlds[t0];           // replicated weights down M
        aw[i].y = wlds[t1];
    }

    const float bias = Bi[0];

    for (int m = 0; m < 16; ++m) {
        int orow = gy0 + ly + m;      // wave-uniform -> EXEC stays all-ones
        if (orow >= H_OUT) break;

        int rbase = (ly + m) * LDSW + lxp;
        v8f acc = {};
#pragma unroll
        for (int i = 0; i < 3; ++i) {
            v2f b;                    // im2col patch: tap t, pixel n
            b.x = tile[rbase + off[i][0]];
            b.y = tile[rbase + off[i][1]];
            // D = A(16x4 weights) x B(4x16 patches) + C ; all rows of D == out[n]
            acc = __builtin_amdgcn_wmma_f32_16x16x4_f32(
                false, aw[i], false, b, (short)0, acc, false, false);
        }

        int ocol = gx0 + lxp + n;
        if (half == 0 && ocol < W_OUT)        // lanes 0-15: coalesced 64B row
            out[(size_t)orow * W_OUT + ocol] = acc[0] + bias;
    }
}

extern "C" void kernel_launch(void* const* d_in, const int* in_sizes, int n_in,
                              void* d_out, int out_size, void* d_ws, size_t ws_size,
                              hipStream_t stream)
{
    (void)in_sizes; (void)n_in; (void)d_ws; (void)ws_size; (void)out_size;
    const float* X  = (const float*)d_in[0];   // 8192*8192 fp32
    const float* Wt = (const float*)d_in[1];   // 3*3 fp32
    const float* Bi = (const float*)d_in[2];   // 1 fp32
    float* out      = (float*)d_out;           // 8190*8190 fp32

    dim3 grid((W_OUT + TBX - 1) / TBX, (H_OUT + TBY - 1) / TBY);
    dim3 block(256);
    conv3x3_wmma_f32<<<grid, block, 0, stream>>>(X, Wt, Bi, out);
}